// TV_grad_49143015800846
// MI455X (gfx1250) — compile-verified
//
#include <hip/hip_runtime.h>
#include <cstdint>
#include <cstddef>

// TV-flow divergence: d = div( grad(x) / sqrt(1 + |grad(x)|^2) )
// Shapes: x, d : (8,4,1024,1024) f32 -> 32 independent 1024x1024 slices.
// Memory-bound (AI ~3 flop/byte). One fused pass through LDS:
//   Stage A: async-DMA (global_load_async_to_lds_b32) a 36x36 clamped halo tile of x into LDS
//   Stage B: compute u0,u1 on 34x34 (output tile + halo 1) in LDS
//   Stage C: divergence on 32x32, streamed out with non-temporal stores.

#define TILE 32        // output tile edge
#define XT   36        // x tile edge (halo 2)
#define XTS  37        // padded LDS stride (odd -> fewer bank conflicts)
#define UT   34        // u tile edge (halo 1)
#define UTS  35

__global__ __launch_bounds__(256) void tv_div_kernel(const float* __restrict__ x,
                                                     float* __restrict__ out,
                                                     int H, int W) {
    __shared__ float lx [XT * XTS];
    __shared__ float lu0[UT * UTS];
    __shared__ float lu1[UT * UTS];

    const int tid   = threadIdx.x;
    const int row0  = blockIdx.y * TILE;   // first output row of this tile
    const int col0  = blockIdx.x * TILE;   // first output col of this tile
    const size_t slice_off = (size_t)blockIdx.z * (size_t)H * (size_t)W;
    const float* xs = x   + slice_off;
    float*       os = out + slice_off;

    // ---------------- Stage A: async global -> LDS tile load (clamped halo-2) ----------------
    for (int i = tid; i < XT * XT; i += 256) {
        const int r = i / XT;
        const int c = i - r * XT;
        int gr = row0 + r - 2; gr = gr < 0 ? 0 : (gr >= H ? H - 1 : gr);
        int gc = col0 + c - 2; gc = gc < 0 ? 0 : (gc >= W ? W - 1 : gc);
        const float* src = xs + (size_t)gr * W + gc;
        // Low 32 bits of a flat LDS address are the byte offset from the wave's LDS base,
        // which is exactly what VDST of the async-LDS load expects.
        unsigned lds_off = (unsigned)(size_t)(&lx[r * XTS + c]);
        unsigned long long gaddr = (unsigned long long)(size_t)src;
        asm volatile("global_load_async_to_lds_b32 %0, %1, off"
                     :
                     : "v"(lds_off), "v"(gaddr)
                     : "memory");
    }
    // Wait for this wave's async DMA (ASYNCcnt), then cross-wave barrier.
    asm volatile("s_wait_asynccnt 0" ::: "memory");
    __syncthreads();

    // ---------------- Stage B: u0,u1 on output tile + halo 1 ----------------
    for (int i = tid; i < UT * UT; i += 256) {
        const int r = i / UT;
        const int c = i - r * UT;
        const int gi = row0 + r - 1;       // global row of this u point (may be -1 or H; then unused)
        const int gj = col0 + c - 1;
        const int rl = r + 1, cl = c + 1;  // u(r,c) <-> lx(r+1, c+1)

        const float xm = lx[(rl - 1) * XTS + cl];
        const float xp = lx[(rl + 1) * XTS + cl];
        const float ym = lx[rl * XTS + cl - 1];
        const float yp = lx[rl * XTS + cl + 1];
        // jnp.gradient scale: 0.5 interior, 1.0 one-sided at edges (loads already clamped)
        const float si = (gi > 0 && gi < H - 1) ? 0.5f : 1.0f;
        const float sj = (gj > 0 && gj < W - 1) ? 0.5f : 1.0f;
        const float gx = (xp - xm) * si;
        const float gy = (yp - ym) * sj;
        const float inorm = 1.0f / sqrtf(1.0f + gx * gx + gy * gy);  // EPSILON = 1.0
        lu0[r * UTS + c] = gx * inorm;
        lu1[r * UTS + c] = gy * inorm;
    }
    __syncthreads();

    // ---------------- Stage C: divergence on the 32x32 interior ----------------
    for (int i = tid; i < TILE * TILE; i += 256) {
        const int r = i / TILE;
        const int c = i - r * TILE;
        const int gi = row0 + r;
        const int gj = col0 + c;
        // u(global g) lives at local index g - (row0-1) => output (r,c) <-> u local (r+1, c+1)
        const int rip = (gi < H - 1) ? r + 2 : r + 1;   // clamped +1 neighbor (rows)
        const int rim = (gi > 0)     ? r     : r + 1;   // clamped -1 neighbor (rows)
        const int cjp = (gj < W - 1) ? c + 2 : c + 1;   // clamped +1 neighbor (cols)
        const int cjm = (gj > 0)     ? c     : c + 1;   // clamped -1 neighbor (cols)
        const float si = (gi > 0 && gi < H - 1) ? 0.5f : 1.0f;
        const float sj = (gj > 0 && gj < W - 1) ? 0.5f : 1.0f;

        const float d = (lu0[rip * UTS + (c + 1)] - lu0[rim * UTS + (c + 1)]) * si
                      + (lu1[(r + 1) * UTS + cjp] - lu1[(r + 1) * UTS + cjm]) * sj;

        // Output is streamed once and never re-read: bypass cache retention.
        __builtin_nontemporal_store(d, &os[(size_t)gi * W + gj]);
    }
}

extern "C" void kernel_launch(void* const* d_in, const int* in_sizes, int n_in,
                              void* d_out, int out_size, void* d_ws, size_t ws_size,
                              hipStream_t stream) {
    (void)n_in; (void)d_ws; (void)ws_size; (void)out_size;
    const float* x = (const float*)d_in[0];
    float* out = (float*)d_out;

    const int H = 1024, W = 1024;
    const int BC = in_sizes[0] / (H * W);   // 8*4 = 32 slices

    dim3 grid(W / TILE, H / TILE, BC);      // (32, 32, 32)
    dim3 block(256);                        // 8 wave32 waves per WGP workgroup
    tv_div_kernel<<<grid, block, 0, stream>>>(x, out, H, W);
}